// SpeicalSpspmm_62706522522386
// MI455X (gfx1250) — compile-verified
//
#include <hip/hip_runtime.h>

// ---------------------------------------------------------------------------
// Sparse(COO) x Sparse(COO) -> Dense f32, N = 4096.
// Densify B in workspace (64 MB, L2-resident in MI455X's 192 MB L2), build a
// row-sorted CSR of A with per-16-row-block padding to a multiple of 4, then
// feed A's nonzeros 4-at-a-time as the K dimension of V_WMMA_F32_16X16X4_F32
// against gathered dense-B rows. Branch-free inner loop (padding replaces
// tail masking), 4 N-tiles per wave, 32-bit GVS addressing, exact f32 accum.
// ---------------------------------------------------------------------------

#define N_DIM 4096
#define NBLK  (N_DIM / 16)   // 256 row-blocks

typedef float v2f __attribute__((ext_vector_type(2)));
typedef float v8f __attribute__((ext_vector_type(8)));

// ---- zero [Bdense | row_count | row_fill | ent] region
__global__ __launch_bounds__(256) void k_zero(uint4* __restrict__ p, int n4) {
  int i = blockIdx.x * blockDim.x + threadIdx.x;
  int stride = gridDim.x * blockDim.x;
  uint4 z = make_uint4(0u, 0u, 0u, 0u);
  for (; i < n4; i += stride) p[i] = z;
}

// ---- scatter-densify B (duplicate coords sum, matching coalesce semantics)
__global__ __launch_bounds__(256) void k_scatter_b(const int* __restrict__ rows,
                                                   const int* __restrict__ cols,
                                                   const float* __restrict__ vals,
                                                   float* __restrict__ Bd, int nnz) {
  int e = blockIdx.x * blockDim.x + threadIdx.x;
  if (e >= nnz) return;
  atomicAdd(&Bd[rows[e] * N_DIM + cols[e]], vals[e]);
}

// ---- histogram A rows
__global__ __launch_bounds__(256) void k_count_a(const int* __restrict__ rows,
                                                 int* __restrict__ cnt, int nnz) {
  int e = blockIdx.x * blockDim.x + threadIdx.x;
  if (e < nnz) atomicAdd(&cnt[rows[e]], 1);
}

// ---- single-block scan: per-row offsets + per-16-row-block padded pointers.
// Thread b (0..255) owns block b: rows 16b..16b+15. Block regions are padded
// to a multiple of 4 entries so the WMMA loop needs no tail masking.
__global__ __launch_bounds__(256) void k_scan(const int* __restrict__ cnt,
                                              int* __restrict__ rptr,
                                              int* __restrict__ bptr) {
  __shared__ int s[256];
  int b = threadIdx.x;
  int c[16];
  int sum = 0;
#pragma unroll
  for (int i = 0; i < 16; ++i) { c[i] = cnt[b * 16 + i]; sum += c[i]; }
  int padded = (sum + 3) & ~3;
  s[b] = padded;
  __syncthreads();
  for (int off = 1; off < 256; off <<= 1) {
    int v = (b >= off) ? s[b - off] : 0;
    __syncthreads();
    s[b] += v;
    __syncthreads();
  }
  int base = s[b] - padded;
  bptr[b] = base;
  if (b == 255) bptr[256] = s[255];
  int run = base;
#pragma unroll
  for (int i = 0; i < 16; ++i) { rptr[b * 16 + i] = run; run += c[i]; }
}

// ---- scatter A entries into padded row-sorted CSR.
// ent.x packs (col<<4 | row&15); ent.y is the f32 value's bits.
__global__ __launch_bounds__(256) void k_fill_a(const int* __restrict__ rows,
                                                const int* __restrict__ cols,
                                                const float* __restrict__ vals,
                                                const int* __restrict__ rptr,
                                                int* __restrict__ cursor,
                                                int2* __restrict__ ent, int nnz) {
  int e = blockIdx.x * blockDim.x + threadIdx.x;
  if (e >= nnz) return;
  int r = rows[e];
  int pos = rptr[r] + atomicAdd(&cursor[r], 1);
  ent[pos] = make_int2((cols[e] << 4) | (r & 15), __float_as_int(vals[e]));
}

// ---- WMMA SpMM: one wave owns four 16-col tiles of a 16-row C block.
// A-operand (16x4 f32): lanes 0-15 hold K=0/1 in v0/v1; lanes 16-31 hold K=2/3.
// B-operand (4x16 f32): v0 = rows {K0 (lo lanes), K2 (hi lanes)}, v1 = {K1, K3}.
// C/D: VGPR g -> M=g (lanes 0-15) / M=g+8 (lanes 16-31), N = lane&15.
__global__ __launch_bounds__(256) void k_spmm_wmma(const int* __restrict__ bptr,
                                                   const int2* __restrict__ ent,
                                                   const float* __restrict__ Bd,
                                                   float* __restrict__ C) {
  const int lane = threadIdx.x & 31;
  const int wave = threadIdx.x >> 5;
  const int half = lane >> 4;      // 0: entries K=0,1   1: entries K=2,3
  const int ln   = lane & 15;      // M for A, N-within-tile for B/C
  const int rb   = blockIdx.y;     // 16-row block of A/C
  const int colbase = (blockIdx.x * 8 + wave) * 64;  // four 16-col tiles

  const int start = bptr[rb];      // padded region: (end-start) % 4 == 0
  const int end   = bptr[rb + 1];

  v8f acc0 = {0.f, 0.f, 0.f, 0.f, 0.f, 0.f, 0.f, 0.f};
  v8f acc1 = acc0, acc2 = acc0, acc3 = acc0;

  const int col = colbase + ln;    // 32-bit element offsets (GVS + scale_offset)

  for (int g = start; g < end; g += 4) {
    // Branch-free: padding entries are (meta=0, val=0) -> contribute exactly 0.
    int ei = g + half * 2;
    int2 E0 = ent[ei];
    int2 E1 = ent[ei + 1];
    int   k0 = E0.x >> 4, r0 = E0.x & 15;
    int   k1 = E1.x >> 4, r1 = E1.x & 15;
    float v0 = __int_as_float(E0.y);
    float v1 = __int_as_float(E1.y);

    // Sparse-scatter A tile: column j has value v_j at row r_j, else 0.
    v2f a;
    a.x = (ln == r0) ? v0 : 0.0f;
    a.y = (ln == r1) ? v1 : 0.0f;

    // Gather 2 dense-B rows (L2-resident) across the four owned N-tiles.
    int off0 = k0 * N_DIM + col;   // < 2^24, 32-bit offset addressing
    int off1 = k1 * N_DIM + col;
    v2f b0, b1, b2, b3;
    b0.x = Bd[off0];      b0.y = Bd[off1];
    b1.x = Bd[off0 + 16]; b1.y = Bd[off1 + 16];
    b2.x = Bd[off0 + 32]; b2.y = Bd[off1 + 32];
    b3.x = Bd[off0 + 48]; b3.y = Bd[off1 + 48];

    acc0 = __builtin_amdgcn_wmma_f32_16x16x4_f32(false, a, false, b0,
                                                 (short)0, acc0, false, false);
    acc1 = __builtin_amdgcn_wmma_f32_16x16x4_f32(false, a, false, b1,
                                                 (short)0, acc1, false, false);
    acc2 = __builtin_amdgcn_wmma_f32_16x16x4_f32(false, a, false, b2,
                                                 (short)0, acc2, false, false);
    acc3 = __builtin_amdgcn_wmma_f32_16x16x4_f32(false, a, false, b3,
                                                 (short)0, acc3, false, false);
  }

  // Write C tiles (every element of d_out written exactly once).
  const int cbase = rb * 16 * N_DIM + col;   // < 2^26, 32-bit offsets
#pragma unroll
  for (int g8 = 0; g8 < 8; ++g8) {
    int rowoff = cbase + (g8 + half * 8) * N_DIM;
    C[rowoff]      = acc0[g8];
    C[rowoff + 16] = acc1[g8];
    C[rowoff + 32] = acc2[g8];
    C[rowoff + 48] = acc3[g8];
  }
}

extern "C" void kernel_launch(void* const* d_in, const int* in_sizes, int n_in,
                              void* d_out, int out_size, void* d_ws, size_t ws_size,
                              hipStream_t stream) {
  const int*   a_idx = (const int*)d_in[0];    // (2, nnzA) row-major
  const float* a_val = (const float*)d_in[1];
  const int*   b_idx = (const int*)d_in[2];    // (2, nnzB)
  const float* b_val = (const float*)d_in[3];
  const int nnzA = in_sizes[1];
  const int nnzB = in_sizes[3];
  float* C = (float*)d_out;

  // Workspace layout (zeroed prefix, then scan outputs):
  //   [ Bdense f32[N*N] | row_count i32[4096] | row_fill i32[4096] |
  //     ent int2[entCap] ]  <- zeroed (padding slots must be 0)
  //   [ row_ptr i32[4096] | blk_ptr i32[257] ]
  const int entCap = ((nnzA + 3) & ~3) + NBLK * 4;  // worst-case padding slack
  char* ws = (char*)d_ws;
  float* Bd        = (float*)ws;
  int*   row_count = (int*)(ws + (size_t)N_DIM * N_DIM * sizeof(float));
  int*   row_fill  = row_count + N_DIM;
  int2*  ent       = (int2*)(row_fill + N_DIM);
  int*   row_ptr   = (int*)(ent + entCap);
  int*   blk_ptr   = row_ptr + N_DIM;

  int zero_dwords = N_DIM * N_DIM + 2 * N_DIM + 2 * entCap;
  zero_dwords = (zero_dwords + 3) & ~3;  // overshoot lands in row_ptr, rewritten later
  k_zero<<<2048, 256, 0, stream>>>((uint4*)ws, zero_dwords / 4);

  k_scatter_b<<<(nnzB + 255) / 256, 256, 0, stream>>>(b_idx, b_idx + nnzB, b_val,
                                                      Bd, nnzB);
  k_count_a<<<(nnzA + 255) / 256, 256, 0, stream>>>(a_idx, row_count, nnzA);
  k_scan<<<1, 256, 0, stream>>>(row_count, row_ptr, blk_ptr);
  k_fill_a<<<(nnzA + 255) / 256, 256, 0, stream>>>(a_idx, a_idx + nnzA, a_val,
                                                   row_ptr, row_fill, ent, nnzA);

  dim3 grid(8, NBLK);  // 8 col-groups x 256 row-blocks; 8 waves x 64 cols each
  k_spmm_wmma<<<grid, 256, 0, stream>>>(blk_ptr, ent, Bd, C);
}